// SpiralConv_54606214201418
// MI455X (gfx1250) — compile-verified
//
#include <hip/hip_runtime.h>

typedef __attribute__((ext_vector_type(2))) float v2f;
typedef __attribute__((ext_vector_type(8))) float v8f;
typedef __attribute__((ext_vector_type(4))) unsigned int v4u;
typedef __attribute__((ext_vector_type(8))) int v8i;
typedef __attribute__((ext_vector_type(4))) int v4i;

#define V_DIM 20000
#define B_DIM 16
#define C_DIM 64
#define L_DIM 9
#define O_DIM 64
#define K_DIM 576          // C*L
#define BC    1024         // B*C

#define SW_STRIDE   580    // 576 + 4 pad -> A-operand ds_load_b64 conflict-free
#define SB_BSTR     68     // per-batch stride (64 + 4 pad) inside a j-row
#define SB_ROW      1088   // 16 batches * 68 = padded j-row (TDM pad produces this)
#define SB_SIZE     (L_DIM * SB_ROW)      // 9792 floats per v-tile
#define SB_BYTES    (SB_SIZE * 4)         // 39168 B per v-tile
#define SOUT_BSTR   1028   // 64*16 + 4 pad per batch row

// ---------------------------------------------------------------------------
// Pass 1: transpose x (BC=1024 rows, V cols) -> xT (V rows, 1024 cols)
// so each gathered column is one contiguous 4KB block (L2-resident: 82MB<192MB)
// ---------------------------------------------------------------------------
__global__ __launch_bounds__(256) void transpose_kernel(
    const float* __restrict__ x, float* __restrict__ xT) {
  __shared__ float tile[32][33];
  const int v0 = blockIdx.x * 32;          // V/32 = 625 exact
  const int r0 = blockIdx.y * 32;          // BC/32 = 32 exact
  const int tx = threadIdx.x, ty = threadIdx.y;
#pragma unroll
  for (int i = 0; i < 4; ++i)
    tile[ty + 8 * i][tx] = x[(size_t)(r0 + ty + 8 * i) * V_DIM + v0 + tx];
  __syncthreads();
#pragma unroll
  for (int i = 0; i < 4; ++i)
    xT[(size_t)(v0 + ty + 8 * i) * BC + r0 + tx] = tile[tx][ty + 8 * i];
}

// ---------------------------------------------------------------------------
// TDM gather (one per v): 9 indexed rows of 1024 f32 from xT -> LDS, with
// hardware padding of 4 DWORDs every 64 DWORDs => LDS layout [j][b*68 + c],
// j-rows at stride 1088 floats.  D# packed per CDNA5 ISA.
// ---------------------------------------------------------------------------
__device__ __forceinline__ void tdm_gather9(const int* __restrict__ sidx,
                                            int v, unsigned lds_byte,
                                            const float* __restrict__ xT) {
  unsigned idx[L_DIM];
#pragma unroll
  for (int j = 0; j < L_DIM; ++j)
    idx[j] = (unsigned)__builtin_amdgcn_readfirstlane(sidx[j * V_DIM + v]) & 0xffffu;

  const unsigned long long ga = (unsigned long long)(uintptr_t)xT;

  v4u g0;
  g0.x = 0x80000001u;                         // count=1 | gather_mode=1 | 16b idx
  g0.y = lds_byte;                            // LDS byte address of tile start
  g0.z = (unsigned)(ga & 0xffffffffu);        // global_addr[31:0]
  g0.w = (unsigned)((ga >> 32) & 0x01ffffffu) // global_addr[56:32]
         | (2u << 30);                        // type = 2 ("image")

  v8i g1;
  g1[0] = (int)((2u << 16)      // data_size = 4 bytes
              | (1u << 20)      // pad_enable (load path)
              | (5u << 22)      // pad_interval: every 64 DWORDs
              | (3u << 25));    // pad_amount: 4 DWORDs  -> b-stride 68 floats
  g1[1] = (int)(1024u << 16);   // tensor_dim0[15:0] @ bits 63:48
  g1[2] = (int)((unsigned)V_DIM << 16);  // tensor_dim1[15:0] @ bits 95:80
  g1[3] = (int)(1024u << 16);   // tile_dim0 = 1024 @ bits 127:112
  g1[4] = L_DIM;                // tile_dim1 = #valid gather indices (9)
  g1[5] = 1024;                 // tensor_dim0_stride[31:0] (row pitch, elems)
  g1[6] = 0;
  g1[7] = 0;

  v4i g2, g3;                   // 16-bit row indices 0..8
  g2.x = (int)(idx[0] | (idx[1] << 16));
  g2.y = (int)(idx[2] | (idx[3] << 16));
  g2.z = (int)(idx[4] | (idx[5] << 16));
  g2.w = (int)(idx[6] | (idx[7] << 16));
  g3.x = (int)idx[8];
  g3.y = 0; g3.z = 0; g3.w = 0;

  v8i g4 = {0, 0, 0, 0, 0, 0, 0, 0};  // reserved/VADDR4 group (zero-filled)

  __builtin_amdgcn_tensor_load_to_lds(g0, g1, g2, g3, g4, /*cpol=*/0);
}

// ---------------------------------------------------------------------------
// Pass 2: per-v GEMM  out[:, :, v] = W(64x576) * neigh_v(576x16)  via
// V_WMMA_F32_16X16X4_F32.  4 waves (one per 16-row M tile), 1 v per iteration,
// TDM double-buffered: tile(it+1) streams in while tile(it) is computed
// (TENSORcnt is in-order per wave, so s_wait_tensorcnt(1) = tile(it) ready).
// K is reordered as k' = c + 64*j (same permutation applied to W in LDS), so
// within a j-row both operands advance linearly: A and B fetches become
// ds_load_2addr_b64 with immediate offsets, conflict-free.
// neigh[b, 9c+j, v] = xT[ S[j*V+v] ][ b*64+c ]   (S = flat spiral_idx)
// Bias omitted: it cancels exactly in the batch-norm.
// ---------------------------------------------------------------------------
__global__ __launch_bounds__(128) void spiral_wmma_kernel(
    const float* __restrict__ xT, const int* __restrict__ sidx,
    const float* __restrict__ W, float* __restrict__ out,
    float* __restrict__ stats) {
  __shared__ float sW[O_DIM * SW_STRIDE];     // 148480 B
  __shared__ float sB[2 * SB_SIZE];           //  78336 B (2 ping-pong v-tiles)
  __shared__ float sOut[B_DIM * SOUT_BSTR];   //  65792 B

  const int tid  = threadIdx.x;
  const int wave = tid >> 5;     // 0..3 = M tile (o-range wave*16 .. +15)
  const int lane = tid & 31;
  const int lo   = lane & 15;    // N index (batch b) / M row within tile
  const int hi   = lane >> 4;    // half-wave: K pair select / +8 M rows in D

  const unsigned sb_byte = (unsigned)(uintptr_t)&sB[0];
  const int vbase = blockIdx.x * 16;   // 1250 blocks * 16 v = 20000

  // ---- prologue: start DMA of tile 0 while W is being staged
  if (wave == 0) tdm_gather9(sidx, vbase, sb_byte, xT);

  // ---- stage W into LDS with the K permutation: col (c + 64j) <- W[o][9c+j]
  {
    const float4* W4 = (const float4*)W;
    for (int i = tid; i < O_DIM * (K_DIM / 4); i += 128) {
      const int o = i / (K_DIM / 4);
      const int q = i - o * (K_DIM / 4);
      const float4 w = W4[i];
      const float vals[4] = {w.x, w.y, w.z, w.w};
#pragma unroll
      for (int e = 0; e < 4; ++e) {
        const int m = 4 * q + e;
        const int c = m / 9, j = m - 9 * c;
        sW[o * SW_STRIDE + c + 64 * j] = vals[e];
      }
    }
  }

  for (int it = 0; it < 16; ++it) {
    // ---- prefetch tile(it+1) into the other buffer; ensure tile(it) landed
    if (wave == 0) {
      if (it < 15) {
        tdm_gather9(sidx, vbase + it + 1,
                    sb_byte + (unsigned)((it + 1) & 1) * SB_BYTES, xT);
        __builtin_amdgcn_s_wait_tensorcnt(1);   // oldest (tile it) complete
      } else {
        __builtin_amdgcn_s_wait_tensorcnt(0);
      }
    }
    __syncthreads();

    // ---- compute: D(16x16) = A(16x576) x B(576x16); j-rows of 64 K' each.
    // Inside a row both lanes' K-pairs stay in the row, so addresses are
    // linear -> constant-offset 2addr b64 LDS loads, ~zero address VALU.
    v8f acc = {0.f, 0.f, 0.f, 0.f, 0.f, 0.f, 0.f, 0.f};
    const float* sWrow = &sW[(wave * 16 + lo) * SW_STRIDE + 2 * hi];
    const float* sBrow = &sB[(it & 1) * SB_SIZE + lo * SB_BSTR + 2 * hi];
    for (int j = 0; j < L_DIM; ++j) {
      const float* aj = sWrow + 64 * j;
      const float* bj = sBrow + SB_ROW * j;
#pragma unroll
      for (int c4 = 0; c4 < 16; ++c4) {
        v2f a  = *(const v2f*)&aj[4 * c4];   // A: K' = 64j+4c4+2hi, +1
        v2f bm = *(const v2f*)&bj[4 * c4];   // B: same K' pair, N = lo
        acc = __builtin_amdgcn_wmma_f32_16x16x4_f32(
            false, a, false, bm, (short)0, acc, false, false);
      }
    }

    // ---- D tile -> sOut staging (lane lo = batch b; vgpr r -> o row)
#pragma unroll
    for (int r = 0; r < 8; ++r) {
      const int o = wave * 16 + r + 8 * hi;
      sOut[lo * SOUT_BSTR + o * 16 + it] = acc[r];
    }
    __syncthreads();   // compute(it) fully retired before DMA reuses buffer
  }

  // ---- coalesced write-out of 16 v's + per-channel sum / sumsq
  const int k  = tid & 3;        // v quad within the 16-v window
  const int ob = tid >> 2;       // o base: 0..31; this thread covers ob, ob+32
  float s1[2] = {0.f, 0.f}, s2[2] = {0.f, 0.f};
  for (int s = 0; s < 32; ++s) {
    const int i = tid + 128 * s;
    const int o = (i >> 2) & 63;         // = ob (s even) or ob+32 (s odd)
    const int b = i >> 8;                // = s/2
    const float4 vv = *(const float4*)&sOut[b * SOUT_BSTR + o * 16 + 4 * k];
    *(float4*)&out[(size_t)(b * 64 + o) * V_DIM + vbase + 4 * k] = vv;
    s1[s & 1] += vv.x + vv.y + vv.z + vv.w;
    s2[s & 1] += vv.x * vv.x + vv.y * vv.y + vv.z * vv.z + vv.w * vv.w;
  }
#pragma unroll
  for (int h = 0; h < 2; ++h) {
    s1[h] += __shfl_down(s1[h], 2, 4);
    s1[h] += __shfl_down(s1[h], 1, 4);
    s2[h] += __shfl_down(s2[h], 2, 4);
    s2[h] += __shfl_down(s2[h], 1, 4);
  }
  if (k == 0) {
    atomicAdd(&stats[ob], s1[0]);
    atomicAdd(&stats[64 + ob], s2[0]);
    atomicAdd(&stats[ob + 32], s1[1]);
    atomicAdd(&stats[96 + ob], s2[1]);
  }
}

// ---------------------------------------------------------------------------
// Pass 3: batch-norm (stats over B*V per channel) + ReLU, in place on d_out.
// ---------------------------------------------------------------------------
__global__ __launch_bounds__(256) void bn_relu_kernel(
    float* __restrict__ out, const float* __restrict__ stats,
    const float* __restrict__ gamma, const float* __restrict__ beta) {
  const float invN = 1.0f / (float)(B_DIM * V_DIM);
  const size_t n4 = (size_t)B_DIM * O_DIM * V_DIM / 4;
  for (size_t i = blockIdx.x * (size_t)blockDim.x + threadIdx.x; i < n4;
       i += (size_t)gridDim.x * blockDim.x) {
    const size_t flat = i * 4;
    const int o = (int)((flat / V_DIM) & 63);
    const float mean = stats[o] * invN;
    const float var  = stats[64 + o] * invN - mean * mean;
    const float inv  = rsqrtf(var + 1e-5f);
    const float g = gamma[o], bb = beta[o];
    float4 v = *(float4*)&out[flat];
    v.x = fmaxf(g * (v.x - mean) * inv + bb, 0.f);
    v.y = fmaxf(g * (v.y - mean) * inv + bb, 0.f);
    v.z = fmaxf(g * (v.z - mean) * inv + bb, 0.f);
    v.w = fmaxf(g * (v.w - mean) * inv + bb, 0.f);
    *(float4*)&out[flat] = v;
  }
}

// ---------------------------------------------------------------------------
extern "C" void kernel_launch(void* const* d_in, const int* in_sizes, int n_in,
                              void* d_out, int out_size, void* d_ws,
                              size_t ws_size, hipStream_t stream) {
  const float* x     = (const float*)d_in[0];   // (16, 64, 20000) f32
  const int*   sidx  = (const int*)d_in[1];     // (20000, 9) i32 (used flat)
  const float* W     = (const float*)d_in[2];   // (64, 576) f32
  // d_in[3] = bias: cancels exactly in batch-norm, unused.
  const float* gamma = (const float*)d_in[4];
  const float* beta  = (const float*)d_in[5];
  float* out   = (float*)d_out;
  float* stats = (float*)d_ws;                  // 128 floats (sum, sumsq)
  float* xT    = stats + 256;                   // 20000 * 1024 floats

  (void)hipMemsetAsync(d_ws, 0, 1024, stream);
  transpose_kernel<<<dim3(V_DIM / 32, BC / 32), dim3(32, 8), 0, stream>>>(x, xT);
  spiral_wmma_kernel<<<V_DIM / 16, 128, 0, stream>>>(xT, sidx, W, out, stats);
  bn_relu_kernel<<<2048, 256, 0, stream>>>(out, stats, gamma, beta);
}